// ext_mem_46531675685183
// MI455X (gfx1250) — compile-verified
//
#include <hip/hip_runtime.h>
#include <math.h>

// Problem constants (match reference)
#define BB 128
#define TT 2048
#define DD 128
#define MM 64
#define BT (BB * TT)   // 262144 rows

typedef __attribute__((ext_vector_type(2))) float v2f;
typedef __attribute__((ext_vector_type(4))) float v4f;
typedef __attribute__((ext_vector_type(8))) float v8f;
typedef __attribute__((ext_vector_type(4))) unsigned int u32x4;
typedef __attribute__((ext_vector_type(4))) int i32x4;
typedef __attribute__((ext_vector_type(8))) int i32x8;

// ---------------------------------------------------------------------------
// TDM helper: async 2-D tile DMA, global -> LDS (CDNA5 Tensor Data Mover).
// D# per CDNA5 ISA:
//   group0: count=1 | lds_addr | global_addr[56:0] | type=2
//   group1: data_size=4B, tensor dims (generous, no OOB), tile dims, stride0
//   groups 2/3 zero (2-D tile: tile_dim2/3 = 0).
// Tracked with TENSORcnt; wait via s_wait_tensorcnt.
// ---------------------------------------------------------------------------
__device__ __forceinline__ unsigned int lds_off(const void* p) {
    // LDS generic address keeps the LDS byte offset in addr[31:0] (ISA §10.2).
    return (unsigned int)(size_t)p;
}

__device__ __forceinline__ void tdm_load_2d(unsigned int lds_addr,
                                            const void* gptr,
                                            unsigned int tile0,   // elems / row
                                            unsigned int tile1,   // rows
                                            unsigned int stride0) // row stride (elems)
{
    unsigned long long ga = (unsigned long long)(size_t)gptr;
    u32x4 g0 = { 1u,                                   // count=1 (valid user D#)
                 lds_addr,
                 (unsigned int)ga,
                 (unsigned int)(ga >> 32) | (2u << 30) };  // type=2 ("image")
    const unsigned int td0 = 1u << 20;                 // tensor_dim0 (no OOB clip)
    const unsigned int td1 = 1u << 20;                 // tensor_dim1
    i32x8 g1 = {
        (int)(2u << 16),                               // data_size=2 -> 4 bytes
        (int)((td0 & 0xffffu) << 16),                  // tensor_dim0[15:0]
        (int)((td0 >> 16) | ((td1 & 0xffffu) << 16)),  // dim0 hi | dim1 lo
        (int)((td1 >> 16) | (tile0 << 16)),            // dim1 hi | tile_dim0
        (int)tile1,                                    // tile_dim1 | tile_dim2=0
        (int)stride0,                                  // tensor_dim0_stride[31:0]
        0,                                             // stride0 hi | d1stride lo
        0 };
    i32x4 z4 = { 0, 0, 0, 0 };
#if defined(__clang_major__) && (__clang_major__ >= 23)
    i32x8 z8 = { 0, 0, 0, 0, 0, 0, 0, 0 };
    __builtin_amdgcn_tensor_load_to_lds(g0, g1, z4, z4, z8, 0);
#else
    __builtin_amdgcn_tensor_load_to_lds(g0, g1, z4, z4, 0);
#endif
}

// ---------------------------------------------------------------------------
// Kernel 1: gate GEMM  out = act(qa @ W^T + b), act = sigmoid (mode 0) / tanh.
// 16 rows x 128 cols per workgroup; 8 waves, one 16x16 fp32-WMMA tile each.
// Branch-free activations (v_exp_f32 + v_rcp_f32), no ocml divergence.
// ---------------------------------------------------------------------------
__global__ void gates_kernel(const float* __restrict__ qa,
                             const float* __restrict__ W,     // [D][D], W[n][k]
                             const float* __restrict__ bias,  // [D]
                             float* __restrict__ outbuf,      // [BT][D]
                             int mode) {
    const int lane = threadIdx.x & 31;
    const int wave = threadIdx.x >> 5;
    const int row0 = blockIdx.x * 16;
    const int n0   = wave * 16;
    const int lr   = lane & 15;
    const int hi   = lane >> 4;

    const float* aptr = qa + (size_t)(row0 + lr) * DD + hi * 2;
    const float* bptr = W  + (size_t)(n0  + lr) * DD + hi * 2;  // B[k][n] = W[n][k]

    v8f c = {};
#pragma unroll 4
    for (int k = 0; k < DD; k += 4) {
        v2f av = *(const v2f*)(aptr + k);
        v2f bv = *(const v2f*)(bptr + k);
        c = __builtin_amdgcn_wmma_f32_16x16x4_f32(
                false, av, false, bv, (short)0, c, false, false);
    }

    const float bb = bias[n0 + lr];
    size_t base = (size_t)(row0 + hi * 8) * DD + (size_t)(n0 + lr);
#pragma unroll
    for (int i = 0; i < 8; ++i) {
        float x = c[i] + bb;
        float y;
        if (mode == 0) {
            y = __builtin_amdgcn_rcpf(1.0f + __expf(-x));           // sigmoid
        } else {
            float z  = __expf(-2.0f * fabsf(x));                    // tanh
            float th = (1.0f - z) * __builtin_amdgcn_rcpf(1.0f + z);
            y = copysignf(th, x);
        }
        outbuf[base + (size_t)i * DD] = y;
    }
}

// ---------------------------------------------------------------------------
// Kernel 2: w = softmax(q @ MS^T) over M=64.
// 16 rows x 64 cols per workgroup (4 waves, one WMMA tile each) + LDS softmax.
// ---------------------------------------------------------------------------
__global__ void wsoftmax_kernel(const float* __restrict__ q,
                                const float* __restrict__ MS,   // [M][D]
                                float* __restrict__ wout) {     // [BT][M]
    __shared__ float sm[16][MM + 8];

    const int lane = threadIdx.x & 31;
    const int wave = threadIdx.x >> 5;
    const int row0 = blockIdx.x * 16;
    const int n0   = wave * 16;
    const int lr   = lane & 15;
    const int hi   = lane >> 4;

    const float* aptr = q  + (size_t)(row0 + lr) * DD + hi * 2;
    const float* bptr = MS + (size_t)(n0  + lr) * DD + hi * 2;

    v8f c = {};
#pragma unroll 4
    for (int k = 0; k < DD; k += 4) {
        v2f av = *(const v2f*)(aptr + k);
        v2f bv = *(const v2f*)(bptr + k);
        c = __builtin_amdgcn_wmma_f32_16x16x4_f32(
                false, av, false, bv, (short)0, c, false, false);
    }

#pragma unroll
    for (int i = 0; i < 8; ++i)
        sm[i + hi * 8][n0 + lr] = c[i];

    __syncthreads();

    if (threadIdx.x < 16) {
        const int r = threadIdx.x;
        float mx = sm[r][0];
#pragma unroll
        for (int m = 1; m < MM; ++m) mx = fmaxf(mx, sm[r][m]);
        float sum = 0.0f;
#pragma unroll
        for (int m = 0; m < MM; ++m) {
            float ex = __expf(sm[r][m] - mx);
            sm[r][m] = ex;
            sum += ex;
        }
        const float inv = __builtin_amdgcn_rcpf(sum);
        float* dst = wout + (size_t)(row0 + r) * MM;
#pragma unroll
        for (int m = 0; m < MM; ++m) dst[m] = sm[r][m] * inv;
    }
}

// ---------------------------------------------------------------------------
// Kernel 3: sequential scan over T, TDM double-buffered.
// Grid = B*2 workgroups (half of D each), 256 threads: 64 cols x 4 lanes.
// Per 32-step chunk, wave 0 issues TDM DMAs (w: 2048 contiguous floats;
// e/a: 64x32 strided tiles) into the back LDS buffer while all waves compute
// the front buffer. s_wait_tensorcnt + barrier between chunks.
// ---------------------------------------------------------------------------
#define CH 32                 // timesteps per chunk
#define NCH (TT / CH)         // 64 chunks

__global__ void scan_kernel(const float* __restrict__ w_ws,   // [B][T][M]
                            const float* __restrict__ e_ws,   // [B][T][D]
                            const float* __restrict__ a_ws,   // [B][T][D]
                            const float* __restrict__ dy_mem, // [M][D]
                            float* __restrict__ out) {        // [B][T][D]
    __shared__ float wbuf[2][CH * MM];   // 2 x 8 KB
    __shared__ float ebuf[2][CH * 64];   // 2 x 8 KB
    __shared__ float abuf[2][CH * 64];   // 2 x 8 KB

    const int b   = blockIdx.x >> 1;
    const int h   = blockIdx.x & 1;
    const int dl  = threadIdx.x >> 2;    // 0..63 (column within half)
    const int sub = threadIdx.x & 3;     // 4 lanes per column
    const int d   = h * 64 + dl;
    const int m0  = sub * 16;

    float s[16];
#pragma unroll
    for (int j = 0; j < 16; ++j)
        s[j] = dy_mem[(size_t)(m0 + j) * DD + d];

    const float* wg = w_ws + (size_t)b * TT * MM;
    const float* eg = e_ws + (size_t)b * TT * DD + h * 64;
    const float* ag = a_ws + (size_t)b * TT * DD + h * 64;
    float*       op = out  + (size_t)b * TT * DD + d;

    // wave-uniform predicate so waves 1..7 scalar-branch around the TDM issue
    const bool issuer = ((__builtin_amdgcn_readfirstlane(threadIdx.x) >> 5) == 0);

    if (issuer) {
        tdm_load_2d(lds_off(&wbuf[0][0]), wg, CH * MM, 1, CH * MM);
        tdm_load_2d(lds_off(&ebuf[0][0]), eg, 64, CH, DD);
        tdm_load_2d(lds_off(&abuf[0][0]), ag, 64, CH, DD);
    }

    for (int c = 0; c < NCH; ++c) {
        const int buf = c & 1;
        if (issuer) __builtin_amdgcn_s_wait_tensorcnt(0);
        __syncthreads();                 // chunk c resident; back buffer free

        if (issuer && (c + 1 < NCH)) {   // prefetch chunk c+1 into back buffer
            const size_t t1 = (size_t)(c + 1) * CH;
            tdm_load_2d(lds_off(&wbuf[buf ^ 1][0]), wg + t1 * MM, CH * MM, 1, CH * MM);
            tdm_load_2d(lds_off(&ebuf[buf ^ 1][0]), eg + t1 * DD, 64, CH, DD);
            tdm_load_2d(lds_off(&abuf[buf ^ 1][0]), ag + t1 * DD, 64, CH, DD);
        }

        const float* wl = &wbuf[buf][0];
        const float* el = &ebuf[buf][0];
        const float* al = &abuf[buf][0];
#pragma unroll 4
        for (int tl = 0; tl < CH; ++tl) {
            v4f w0 = *(const v4f*)(wl + tl * MM + m0);
            v4f w1 = *(const v4f*)(wl + tl * MM + m0 + 4);
            v4f w2 = *(const v4f*)(wl + tl * MM + m0 + 8);
            v4f w3 = *(const v4f*)(wl + tl * MM + m0 + 12);
            const float ed = el[tl * 64 + dl];
            const float ad = al[tl * 64 + dl];

            float wm[16];
#pragma unroll
            for (int j = 0; j < 4; ++j) {
                wm[j] = w0[j]; wm[4 + j] = w1[j]; wm[8 + j] = w2[j]; wm[12 + j] = w3[j];
            }

            // readout BEFORE the write (reference semantics)
            float pr = 0.0f;
#pragma unroll
            for (int j = 0; j < 16; ++j) pr = fmaf(wm[j], s[j], pr);
            pr += __shfl_xor(pr, 1, 32);
            pr += __shfl_xor(pr, 2, 32);
            if (sub == 0) *op = pr;

            // mem = mem*(1 - w*e) + w*a  ==  mem += w * (a - mem*e)
#pragma unroll
            for (int j = 0; j < 16; ++j) {
                float tmp = fmaf(-s[j], ed, ad);
                s[j] = fmaf(wm[j], tmp, s[j]);
            }
            op += DD;
        }
        __syncthreads();                 // done reading front buffer
    }
}

// ---------------------------------------------------------------------------
extern "C" void kernel_launch(void* const* d_in, const int* in_sizes, int n_in,
                              void* d_out, int out_size, void* d_ws, size_t ws_size,
                              hipStream_t stream) {
    const float* q      = (const float*)d_in[0];
    const float* qa     = (const float*)d_in[1];
    const float* MS     = (const float*)d_in[2];
    const float* dy_mem = (const float*)d_in[3];
    const float* e_w    = (const float*)d_in[4];
    const float* e_b    = (const float*)d_in[5];
    const float* a_w    = (const float*)d_in[6];
    const float* a_b    = (const float*)d_in[7];
    float* out = (float*)d_out;

    // Workspace partition: w (64 MB) | e (128 MB) | a (128 MB)
    char* ws = (char*)d_ws;
    float* w_ws = (float*)ws;
    float* e_ws = (float*)(ws + (size_t)BT * MM * sizeof(float));
    float* a_ws = (float*)(ws + (size_t)BT * MM * sizeof(float)
                              + (size_t)BT * DD * sizeof(float));

    const int row_tiles = BT / 16;   // 16384

    gates_kernel<<<row_tiles, 256, 0, stream>>>(qa, e_w, e_b, e_ws, 0);
    gates_kernel<<<row_tiles, 256, 0, stream>>>(qa, a_w, a_b, a_ws, 1);
    wsoftmax_kernel<<<row_tiles, 128, 0, stream>>>(q, MS, w_ws);
    scan_kernel<<<BB * 2, 256, 0, stream>>>(w_ws, e_ws, a_ws, dy_mem, out);
}